// Resampler_66803921322437
// MI455X (gfx1250) — compile-verified
//
#include <hip/hip_runtime.h>
#include <hip/hip_bf16.h>
#include <math.h>

// ---- problem constants (from reference) ----
#define CN   2
#define CH   128
#define CW   128
#define CC   256
#define CG   4
#define CGC  64
#define CP   9
#define CHp  130
#define CWp  130
#define NPIX (CN*CH*CW)          // 32768
#define OMW  108                  // G * 3 * P
#define LDA_S 260                 // padded LDS stride (floats) to avoid bank conflicts

typedef __attribute__((ext_vector_type(2))) float v2f;
typedef __attribute__((ext_vector_type(8))) float v8f;

// workspace layout (bytes)
#define VP_FLOATS ((size_t)CN*CHp*CWp*CC)            // 8,652,800
#define VP_BYTES  (VP_FLOATS*4)                      // 34,611,200 (16B aligned)
#define OM_FLOATS ((size_t)NPIX*OMW)                 // 3,538,944
#define OM_BYTES  (OM_FLOATS*4)                      // 14,155,776 (16B aligned)
#define S_FLOATS  ((size_t)NPIX*CC)

// ----------------------------------------------------------------------------
// Kernel 0: zero fill (padding ring of value buffer; interior overwritten later)
// ----------------------------------------------------------------------------
__global__ __launch_bounds__(256) void k_fill0(float4* __restrict__ p, size_t n4) {
    size_t i = (size_t)blockIdx.x * blockDim.x + threadIdx.x;
    size_t stride = (size_t)gridDim.x * blockDim.x;
    float4 z = make_float4(0.f, 0.f, 0.f, 0.f);
    for (; i < n4; i += stride) p[i] = z;
}

// ----------------------------------------------------------------------------
// Kernel 1: value = sampled_feat @ W_in + b_in, scattered into padded [N,Hp,Wp,C]
//   block = 128 threads (4 waves), one 16-row M tile per block,
//   each wave computes 4 N-tiles of 16 cols. K loop: 64 x v_wmma_f32_16x16x4_f32
// ----------------------------------------------------------------------------
__global__ __launch_bounds__(128) void k_gemm_in(const float* __restrict__ X,
                                                 const float* __restrict__ Wm,
                                                 const float* __restrict__ bias,
                                                 float* __restrict__ Vp) {
    __shared__ float lA[16 * LDA_S];
    const int m0 = blockIdx.x * 16;
    const int t  = threadIdx.x;
    for (int i = t; i < 16 * 256; i += 128) {
        int r = i >> 8, c = i & 255;
        lA[r * LDA_S + c] = X[(size_t)(m0 + r) * CC + c];
    }
    __syncthreads();
    const int wave = t >> 5, lane = t & 31, lr = lane & 15, lh = lane >> 4;
    for (int q = 0; q < 4; ++q) {
        const int n0 = (wave * 4 + q) * 16;
        const int col = n0 + lr;
        v8f acc = {0.f,0.f,0.f,0.f,0.f,0.f,0.f,0.f};
        #pragma unroll 4
        for (int k0 = 0; k0 < 256; k0 += 4) {
            v2f a, b;
            const float* ap = &lA[lr * LDA_S + k0 + 2 * lh];
            a.x = ap[0]; a.y = ap[1];
            const float* bp = &Wm[(size_t)(k0 + 2 * lh) * CC + col];
            b.x = bp[0];  b.y = bp[CC];
            acc = __builtin_amdgcn_wmma_f32_16x16x4_f32(false, a, false, b,
                                                        (short)0, acc, false, false);
        }
        const float bv = bias[col];
        #pragma unroll
        for (int r = 0; r < 8; ++r) {
            int m   = m0 + r + 8 * lh;
            int n   = m >> 14;            // / (H*W)
            int rem = m & 16383;
            int y   = rem >> 7, x = rem & 127;
            Vp[(((size_t)n * CHp + (y + 1)) * CWp + (x + 1)) * CC + col] = acc[r] + bv;
        }
    }
}

// ----------------------------------------------------------------------------
// Kernel 2: ctx = dwconv3x3(context) + dw_bias (into LDS), om = ctx @ W_om + b_om
//   W_om is 256x108; N padded to 112 (7 tiles), guarded loads/stores.
// ----------------------------------------------------------------------------
__global__ __launch_bounds__(128) void k_ctx_om(const float* __restrict__ Cin,
                                                const float* __restrict__ dwk,
                                                const float* __restrict__ dwb,
                                                const float* __restrict__ Wom,
                                                const float* __restrict__ bom,
                                                float* __restrict__ om) {
    __shared__ float lA[16 * LDA_S];
    const int m0 = blockIdx.x * 16;
    const int t  = threadIdx.x;
    for (int i = t; i < 16 * 256; i += 128) {
        int rr = i >> 8, c = i & 255;
        int m = m0 + rr;
        int n = m >> 14, rem = m & 16383, y = rem >> 7, x = rem & 127;
        float s = dwb[c];
        #pragma unroll
        for (int dy = -1; dy <= 1; ++dy) {
            #pragma unroll
            for (int dx = -1; dx <= 1; ++dx) {
                int yy = y + dy, xx = x + dx;
                if (yy >= 0 && yy < CH && xx >= 0 && xx < CW) {
                    s += Cin[(((size_t)n * CH + yy) * CW + xx) * CC + c] *
                         dwk[((dy + 1) * 3 + (dx + 1)) * CC + c];
                }
            }
        }
        lA[rr * LDA_S + c] = s;
    }
    __syncthreads();
    const int wave = t >> 5, lane = t & 31, lr = lane & 15, lh = lane >> 4;
    for (int q = 0; q < 2; ++q) {
        const int nt = wave + q * 4;         // uniform per wave
        if (nt >= 7) break;
        const int n0 = nt * 16;
        const int col = n0 + lr;
        const bool inb = (col < OMW);
        v8f acc = {0.f,0.f,0.f,0.f,0.f,0.f,0.f,0.f};
        #pragma unroll 4
        for (int k0 = 0; k0 < 256; k0 += 4) {
            v2f a, b;
            const float* ap = &lA[lr * LDA_S + k0 + 2 * lh];
            a.x = ap[0]; a.y = ap[1];
            int kb = k0 + 2 * lh;
            b.x = inb ? Wom[(size_t)kb * OMW + col]       : 0.f;
            b.y = inb ? Wom[(size_t)(kb + 1) * OMW + col] : 0.f;
            acc = __builtin_amdgcn_wmma_f32_16x16x4_f32(false, a, false, b,
                                                        (short)0, acc, false, false);
        }
        if (inb) {
            const float bv = bom[col];
            #pragma unroll
            for (int r = 0; r < 8; ++r)
                om[(size_t)(m0 + r + 8 * lh) * OMW + col] = acc[r] + bv;
        }
    }
}

// ----------------------------------------------------------------------------
// Kernel 3: deformable bilinear sampling + mask-weighted tap sum.
//   One wave per (pixel, group); each lane owns 2 channels (float2 loads).
//   value_pad (34.6MB) is L2-resident on MI455X (192MB L2) -> gathers hit L2.
// ----------------------------------------------------------------------------
__global__ __launch_bounds__(256) void k_sample(const float* __restrict__ Vp,
                                                const float* __restrict__ om,
                                                float* __restrict__ S) {
    const int wid  = ((int)blockIdx.x * 256 + (int)threadIdx.x) >> 5;
    const int lane = threadIdx.x & 31;
    const int g    = wid & 3;
    const int pix  = wid >> 2;
    const int n = pix >> 14, rem = pix & 16383, y = rem >> 7, x = rem & 127;

    const float* omr  = om + (size_t)pix * OMW + g * 27;
    const float* vbase = Vp + (size_t)n * CHp * CWp * CC + g * CGC + 2 * lane;

    float a0 = 0.f, a1 = 0.f;
    #pragma unroll
    for (int p = 0; p < CP; ++p) {
        const float ox = omr[2 * p], oy = omr[2 * p + 1], mk = omr[18 + p];
        const float kx = (float)(p % 3) - 1.f;
        const float ky = (float)(p / 3) - 1.f;
        const float px = (float)(x + 1) + 0.5f * (kx + ox);
        const float py = (float)(y + 1) + 0.5f * (ky + oy);
        const float fx = floorf(px), fy = floorf(py);
        const float wx1 = px - fx, wy1 = py - fy;
        const int x0 = (int)fx, y0 = (int)fy;
        const float wts[4] = {(1.f - wx1) * (1.f - wy1), wx1 * (1.f - wy1),
                              (1.f - wx1) * wy1,         wx1 * wy1};
        #pragma unroll
        for (int c4 = 0; c4 < 4; ++c4) {
            const int xi = x0 + (c4 & 1), yi = y0 + (c4 >> 1);
            const bool valid = (xi >= 0) & (xi <= CWp - 1) & (yi >= 0) & (yi <= CHp - 1);
            const int xc = xi < 0 ? 0 : (xi > CWp - 1 ? CWp - 1 : xi);
            const int yc = yi < 0 ? 0 : (yi > CHp - 1 ? CHp - 1 : yi);
            const float w = valid ? (wts[c4] * mk) : 0.f;
            const float2 v = *(const float2*)(vbase + ((size_t)yc * CWp + xc) * CC);
            a0 += v.x * w;
            a1 += v.y * w;
        }
    }
    float2* outp = (float2*)(S + (size_t)pix * CC + g * CGC + 2 * lane);
    *outp = make_float2(a0, a1);
}

// ----------------------------------------------------------------------------
// Kernel 4: out = S @ W_out, then LayerNorm + exact GELU. Fused per 16-row tile.
// ----------------------------------------------------------------------------
__global__ __launch_bounds__(128) void k_out(const float* __restrict__ S,
                                             const float* __restrict__ Wout,
                                             const float* __restrict__ gam,
                                             const float* __restrict__ bet,
                                             float* __restrict__ Out) {
    __shared__ float lA[16 * LDA_S];
    __shared__ float lO[16 * LDA_S];
    __shared__ float red[16][8][2];
    __shared__ float stats[16][2];
    const int m0 = blockIdx.x * 16;
    const int t  = threadIdx.x;
    for (int i = t; i < 16 * 256; i += 128) {
        int r = i >> 8, c = i & 255;
        lA[r * LDA_S + c] = S[(size_t)(m0 + r) * CC + c];
    }
    __syncthreads();
    const int wave = t >> 5, lane = t & 31, lr = lane & 15, lh = lane >> 4;
    for (int q = 0; q < 4; ++q) {
        const int n0 = (wave * 4 + q) * 16;
        const int col = n0 + lr;
        v8f acc = {0.f,0.f,0.f,0.f,0.f,0.f,0.f,0.f};
        #pragma unroll 4
        for (int k0 = 0; k0 < 256; k0 += 4) {
            v2f a, b;
            const float* ap = &lA[lr * LDA_S + k0 + 2 * lh];
            a.x = ap[0]; a.y = ap[1];
            const float* bp = &Wout[(size_t)(k0 + 2 * lh) * CC + col];
            b.x = bp[0];  b.y = bp[CC];
            acc = __builtin_amdgcn_wmma_f32_16x16x4_f32(false, a, false, b,
                                                        (short)0, acc, false, false);
        }
        #pragma unroll
        for (int r = 0; r < 8; ++r)
            lO[(r + 8 * lh) * LDA_S + col] = acc[r];
    }
    __syncthreads();
    // LayerNorm over 256 cols per row: 8 threads/row, 32 elems each
    const int row = t >> 3, seg = t & 7;
    float s = 0.f, ss = 0.f;
    #pragma unroll 8
    for (int j = 0; j < 32; ++j) {
        float v = lO[row * LDA_S + seg * 32 + j];
        s += v; ss += v * v;
    }
    red[row][seg][0] = s; red[row][seg][1] = ss;
    __syncthreads();
    if (seg == 0) {
        float su = 0.f, sq = 0.f;
        #pragma unroll
        for (int j = 0; j < 8; ++j) { su += red[row][j][0]; sq += red[row][j][1]; }
        const float mu  = su * (1.f / 256.f);
        const float var = sq * (1.f / 256.f) - mu * mu;
        stats[row][0] = mu;
        stats[row][1] = rsqrtf(var + 1e-5f);
    }
    __syncthreads();
    const float mu = stats[row][0], rs = stats[row][1];
    #pragma unroll 8
    for (int j = 0; j < 32; ++j) {
        const int c = seg * 32 + j;
        float v = (lO[row * LDA_S + c] - mu) * rs * gam[c] + bet[c];
        float ge = 0.5f * v * (1.f + erff(v * 0.70710678118654752f));
        Out[(size_t)(m0 + row) * CC + c] = ge;
    }
}

// ----------------------------------------------------------------------------
extern "C" void kernel_launch(void* const* d_in, const int* in_sizes, int n_in,
                              void* d_out, int out_size, void* d_ws, size_t ws_size,
                              hipStream_t stream) {
    (void)in_sizes; (void)n_in; (void)out_size; (void)ws_size;
    const float* sampled = (const float*)d_in[0];
    const float* ctx     = (const float*)d_in[1];
    const float* W_in    = (const float*)d_in[2];
    const float* b_in    = (const float*)d_in[3];
    const float* dwk     = (const float*)d_in[4];
    const float* dwb     = (const float*)d_in[5];
    const float* W_om    = (const float*)d_in[6];
    const float* b_om    = (const float*)d_in[7];
    const float* W_out   = (const float*)d_in[8];
    const float* lng     = (const float*)d_in[9];
    const float* lnb     = (const float*)d_in[10];
    float* out = (float*)d_out;

    char* ws = (char*)d_ws;
    float* Vp = (float*)ws;                          // padded value  34.6 MB
    float* om = (float*)(ws + VP_BYTES);             // offset/mask   14.2 MB
    float* Sb = (float*)(ws + VP_BYTES + OM_BYTES);  // sampled feat  33.5 MB

    k_fill0  <<<2048,  256, 0, stream>>>((float4*)Vp, VP_FLOATS / 4);
    k_gemm_in<<<NPIX/16, 128, 0, stream>>>(sampled, W_in, b_in, Vp);
    k_ctx_om <<<NPIX/16, 128, 0, stream>>>(ctx, dwk, dwb, W_om, b_om, om);
    k_sample <<<(NPIX*CG)/8, 256, 0, stream>>>(Vp, om, Sb);
    k_out    <<<NPIX/16, 128, 0, stream>>>(Sb, W_out, lng, lnb, out);
}